// SambaMambaBlock_65687229825697
// MI455X (gfx1250) — compile-verified
//
#include <hip/hip_runtime.h>
#include <hip/hip_bf16.h>

// ---------------------------------------------------------------------------
// Mamba block forward for MI455X (gfx1250, wave32, WMMA + async-LDS copies).
// Pipeline:
//   K1 gemm : xz[2048,4096]   = x[2048,1024] @ W_in[1024,4096]
//   K2 conv : xc[2048,2048]   = silu(depthwise_conv(xz[:, :2048]))
//   K3 gemm : proj[2048,96]   = xc @ W_xproj[2048,96]
//   K4 gemm : delta[2048,2048]= softplus(proj[:, :64] @ W_dt[64,2048] + b_dt)
//   K5 scan : y[2048,2048]    = SSM scan * silu(z)   (z = xz[:, 2048:])
//   K6 gemm : out[2048,1024]  = y @ W_out[2048,1024]
// GEMMs: V_WMMA_F32_16X16X4_F32; tile staging via GLOBAL_LOAD_ASYNC_TO_LDS_B128.
// ---------------------------------------------------------------------------

typedef __attribute__((ext_vector_type(2))) float v2f;
typedef __attribute__((ext_vector_type(8))) float v8f;
typedef int v4i __attribute__((vector_size(16)));   // matches builtin proto

#define AS1 __attribute__((address_space(1)))
#define AS3 __attribute__((address_space(3)))

// Async-copy mode selection:
//   2 = clang builtin, 1 = inline asm (device pass only), 0 = plain fallback
#if defined(__HIP_DEVICE_COMPILE__)
  #if __has_builtin(__builtin_amdgcn_global_load_async_to_lds_b128)
    #define ASYNC_MODE 2
  #else
    #define ASYNC_MODE 1
  #endif
#else
  #define ASYNC_MODE 0
#endif

__device__ __forceinline__ void async_copy_b128(const float* gsrc, float* lds_dst) {
#if ASYNC_MODE == 2
  // AS1 pointers to global memory share value representation with flat ptrs.
  AS1 v4i* g = (AS1 v4i*)(unsigned long long)gsrc;
  // Flat shared addr low 32 bits == LDS byte offset (ISA 10.2 aperture map).
  AS3 v4i* l = (AS3 v4i*)(unsigned int)(unsigned long long)lds_dst;
  __builtin_amdgcn_global_load_async_to_lds_b128(g, l, 0, 0);
#elif ASYNC_MODE == 1
  unsigned int loff = (unsigned int)(unsigned long long)lds_dst;
  asm volatile("global_load_async_to_lds_b128 %0, %1, off"
               :: "v"(loff), "v"(gsrc) : "memory");
#else
  float4 v = *(const float4*)gsrc;
  lds_dst[0] = v.x; lds_dst[1] = v.y; lds_dst[2] = v.z; lds_dst[3] = v.w;
#endif
}

__device__ __forceinline__ void wait_asynccnt0() {
#if ASYNC_MODE == 2
  #if __has_builtin(__builtin_amdgcn_s_wait_asynccnt)
  __builtin_amdgcn_s_wait_asynccnt(0);
  #else
  asm volatile("s_wait_asynccnt 0" ::: "memory");
  #endif
#elif ASYNC_MODE == 1
  asm volatile("s_wait_asynccnt 0" ::: "memory");
#endif
}

namespace cfg {
constexpr int D_MODEL  = 1024;
constexpr int D_STATE  = 16;
constexpr int D_CONV   = 4;
constexpr int D_INNER  = 2048;
constexpr int DT_RANK  = 64;
constexpr int BATCH    = 2;
constexpr int SEQ      = 1024;
constexpr int NTOK     = BATCH * SEQ;               // 2048
constexpr int NPROJ    = DT_RANK + 2 * D_STATE;     // 96
constexpr int XZ_COLS  = 2 * D_INNER;               // 4096
}

// ---------------------------------------------------------------------------
// WMMA f32 tiled GEMM: C[M,N] = A[M,K] @ B[K,N]  (row-major, strided)
// Block: 256 threads = 8 waves, tile 128(M) x 64(N), K staged in chunks of 16.
// Wave w -> (waveRow = w>>1 in 0..3, waveCol = w&1 in 0..1), each wave owns a
// 32x32 patch = 2x2 grid of 16x16 WMMA accumulators.
//
// Fragment layouts per CDNA5 ISA 7.12.2:
//   A 16x4 f32 : lanes 0-15 -> M=lane, regs {K=kk, K=kk+1};
//                lanes 16-31 -> M=lane-16, regs {K=kk+2, K=kk+3}
//   B 4x16 f32 : lanes 0-15 -> N=lane, regs {K=kk, K=kk+1};
//                lanes 16-31 -> N=lane-16, regs {K=kk+2, K=kk+3}
//   C/D 16x16  : reg r, lanes 0-15 -> (M=r,   N=lane),
//                        lanes 16-31 -> (M=8+r, N=lane-16)
// EPI: 0 = none, 1 = softplus(val + bias[col])
// ---------------------------------------------------------------------------
#define GEMM_TM 128
#define GEMM_TN 64
#define GEMM_TK 16
#define AS_STRIDE 20   // 16 + 4 pad: rows 16B aligned (async B128), 20r%64 distinct
#define BS_STRIDE 80   // 64 + 16 pad: rows 16B aligned, half-wave K rows 32 banks apart

template <int EPI>
__global__ __launch_bounds__(256) void wmma_gemm_f32(
    const float* __restrict__ A, const float* __restrict__ B,
    float* __restrict__ C, const float* __restrict__ bias,
    int M, int N, int K, int lda, int ldb, int ldc) {
  __shared__ float As[GEMM_TM * AS_STRIDE];
  __shared__ float Bs[GEMM_TK * BS_STRIDE];

  const int tid     = threadIdx.x;
  const int lane    = tid & 31;
  const int wave    = tid >> 5;
  const int waveRow = wave >> 1;   // 0..3
  const int waveCol = wave & 1;    // 0..1
  const int half    = lane >> 4;   // 0 or 1
  const int l16     = lane & 15;

  const int blockM = blockIdx.y * GEMM_TM;
  const int blockN = blockIdx.x * GEMM_TN;

  // Staging coordinates (fixed per thread).
  const int aRow0 = tid >> 2;              // 0..63   (iter 0)
  const int aC4   = (tid & 3) << 2;        // 0,4,8,12
  const int bRow  = tid >> 4;              // 0..15
  const int bC4   = (tid & 15) << 2;       // 0..60

  const v8f zero8 = {0.f, 0.f, 0.f, 0.f, 0.f, 0.f, 0.f, 0.f};
  v8f acc[2][2];
  acc[0][0] = zero8; acc[0][1] = zero8; acc[1][0] = zero8; acc[1][1] = zero8;

  for (int k0 = 0; k0 < K; k0 += GEMM_TK) {
    // ---- stage A tile: 128 rows x 16 k (M,K multiples of tile: no guards) --
#pragma unroll
    for (int it = 0; it < 2; ++it) {
      int row = aRow0 + it * 64;
      async_copy_b128(A + (size_t)(blockM + row) * lda + k0 + aC4,
                      &As[row * AS_STRIDE + aC4]);
    }
    // ---- stage B tile: 16 k x 64 cols (guard ragged N, e.g. N=96) ---------
    {
      int col = blockN + bC4;
      float* dst = &Bs[bRow * BS_STRIDE + bC4];
      if (col + 4 <= N) {
        async_copy_b128(B + (size_t)(k0 + bRow) * ldb + col, dst);
      } else {
        const float* bp = B + (size_t)(k0 + bRow) * ldb;
        dst[0] = (col + 0 < N) ? bp[col + 0] : 0.f;
        dst[1] = (col + 1 < N) ? bp[col + 1] : 0.f;
        dst[2] = (col + 2 < N) ? bp[col + 2] : 0.f;
        dst[3] = (col + 3 < N) ? bp[col + 3] : 0.f;
      }
    }
    wait_asynccnt0();
    __syncthreads();

    // ---- compute: 4 k-steps of 4, 2x2 WMMAs each --------------------------
#pragma unroll
    for (int kk = 0; kk < GEMM_TK; kk += 4) {
      v2f af[2], bf[2];
#pragma unroll
      for (int i = 0; i < 2; ++i) {
        int r = waveRow * 32 + i * 16 + l16;
        af[i] = *(const v2f*)&As[r * AS_STRIDE + kk + half * 2];
      }
#pragma unroll
      for (int j = 0; j < 2; ++j) {
        int n = waveCol * 32 + j * 16 + l16;
        bf[j].x = Bs[(kk + half * 2 + 0) * BS_STRIDE + n];
        bf[j].y = Bs[(kk + half * 2 + 1) * BS_STRIDE + n];
      }
#pragma unroll
      for (int i = 0; i < 2; ++i)
#pragma unroll
        for (int j = 0; j < 2; ++j)
          acc[i][j] = __builtin_amdgcn_wmma_f32_16x16x4_f32(
              false, af[i], false, bf[j], (short)0, acc[i][j], false, false);
    }
    __syncthreads();
  }

  // ---- epilogue + store ----------------------------------------------------
#pragma unroll
  for (int i = 0; i < 2; ++i) {
#pragma unroll
    for (int j = 0; j < 2; ++j) {
      int col = blockN + waveCol * 32 + j * 16 + l16;
      if (col < N) {
        float bv = (EPI == 1) ? bias[col] : 0.f;
#pragma unroll
        for (int r = 0; r < 8; ++r) {
          int row   = blockM + waveRow * 32 + i * 16 + half * 8 + r;
          float val = acc[i][j][r];
          if (EPI == 1) {
            float v = val + bv;
            val = (v > 20.f) ? v : log1pf(__expf(v));
          }
          C[(size_t)row * ldc + col] = val;
        }
      }
    }
  }
}

// ---------------------------------------------------------------------------
// K2: depthwise causal conv over time + SiLU.
// xz: [NTOK, 4096] (x_b = cols 0..2047). xc: [NTOK, 2048].
// ---------------------------------------------------------------------------
__global__ __launch_bounds__(256) void conv_silu_kernel(
    const float* __restrict__ xz, const float* __restrict__ conv_w,
    const float* __restrict__ conv_b, float* __restrict__ xc) {
  using namespace cfg;
  int idx = blockIdx.x * 256 + threadIdx.x;  // NTOK * D_INNER threads
  int d = idx & (D_INNER - 1);
  int m = idx >> 11;             // token index (D_INNER == 2048)
  int t = m & (SEQ - 1);         // time within batch
  int mb = m - t;                // first token of this batch

  float acc = conv_b[d];
#pragma unroll
  for (int k = 0; k < D_CONV; ++k) {
    int tt = t - (D_CONV - 1) + k;
    if (tt >= 0)
      acc = fmaf(xz[(size_t)(mb + tt) * XZ_COLS + d], conv_w[d * D_CONV + k],
                 acc);
  }
  // silu
  xc[(size_t)m * D_INNER + d] = acc / (1.f + __expf(-acc));
}

// ---------------------------------------------------------------------------
// K5: sequential SSM scan. One lane per (batch, channel): 4096 lanes.
// 16-state recurrence kept in registers; B/C vectors are wave-uniform
// addresses per (b,t) -> broadcast loads from L2.
// Writes y = (scan output) * silu(z).
// ---------------------------------------------------------------------------
__global__ __launch_bounds__(256) void ssm_scan_kernel(
    const float* __restrict__ proj, const float* __restrict__ delta,
    const float* __restrict__ xc, const float* __restrict__ xz,
    const float* __restrict__ A_log, const float* __restrict__ D_param,
    float* __restrict__ y) {
  using namespace cfg;
  int idx = blockIdx.x * 256 + threadIdx.x;  // BATCH * D_INNER
  int d = idx & (D_INNER - 1);
  int b = idx >> 11;

  float Av[D_STATE];
#pragma unroll
  for (int s = 0; s < D_STATE; ++s)
    Av[s] = -__expf(A_log[(size_t)d * D_STATE + s]);
  const float Dp = D_param[d];

  float h[D_STATE];
#pragma unroll
  for (int s = 0; s < D_STATE; ++s) h[s] = 0.f;

  for (int t = 0; t < SEQ; ++t) {
    int m = b * SEQ + t;
    float dly = delta[(size_t)m * D_INNER + d];
    float u   = xc[(size_t)m * D_INNER + d];
    const float* pr = proj + (size_t)m * NPROJ;  // [:64]=dt, [64:80]=B, [80:96]=C
    float acc = 0.f;
#pragma unroll
    for (int s = 0; s < D_STATE; ++s) {
      float Bv = pr[DT_RANK + s];
      float Cv = pr[DT_RANK + D_STATE + s];
      float ab = __expf(dly * Av[s]);
      h[s] = fmaf(ab, h[s], dly * Bv * u);
      acc  = fmaf(h[s], Cv, acc);
    }
    float yv = fmaf(Dp, u, acc);
    float zv = xz[(size_t)m * XZ_COLS + D_INNER + d];
    y[(size_t)m * D_INNER + d] = yv * (zv / (1.f + __expf(-zv)));
  }
}

// ---------------------------------------------------------------------------
extern "C" void kernel_launch(void* const* d_in, const int* in_sizes, int n_in,
                              void* d_out, int out_size, void* d_ws,
                              size_t ws_size, hipStream_t stream) {
  using namespace cfg;
  (void)in_sizes; (void)n_in; (void)out_size; (void)ws_size;

  const float* x       = (const float*)d_in[0];  // [2,1024,1024]
  const float* W_in    = (const float*)d_in[1];  // [1024,4096]
  const float* conv_w  = (const float*)d_in[2];  // [2048,4]
  const float* conv_b  = (const float*)d_in[3];  // [2048]
  const float* W_xproj = (const float*)d_in[4];  // [2048,96]
  const float* W_dt    = (const float*)d_in[5];  // [64,2048]
  const float* b_dt    = (const float*)d_in[6];  // [2048]
  const float* A_log   = (const float*)d_in[7];  // [2048,16]
  const float* D_param = (const float*)d_in[8];  // [2048]
  const float* W_out   = (const float*)d_in[9];  // [2048,1024]
  float* out = (float*)d_out;                    // [2,1024,1024]

  // workspace carve-up (floats)
  float* xz    = (float*)d_ws;                       // 2048*4096
  float* xc    = xz + (size_t)NTOK * XZ_COLS;        // 2048*2048
  float* proj  = xc + (size_t)NTOK * D_INNER;        // 2048*96
  float* delta = proj + (size_t)NTOK * NPROJ;        // 2048*2048
  float* y     = delta + (size_t)NTOK * D_INNER;     // 2048*2048

  // K1: xz = x @ W_in        M=2048 N=4096 K=1024
  {
    dim3 grid(XZ_COLS / GEMM_TN, NTOK / GEMM_TM);
    wmma_gemm_f32<0><<<grid, 256, 0, stream>>>(
        x, W_in, xz, nullptr, NTOK, XZ_COLS, D_MODEL, D_MODEL, XZ_COLS,
        XZ_COLS);
  }
  // K2: xc = silu(conv(x_b))
  {
    int nthreads = NTOK * D_INNER;
    conv_silu_kernel<<<nthreads / 256, 256, 0, stream>>>(xz, conv_w, conv_b,
                                                         xc);
  }
  // K3: proj = xc @ W_xproj  M=2048 N=96 K=2048
  {
    dim3 grid((NPROJ + GEMM_TN - 1) / GEMM_TN, NTOK / GEMM_TM);
    wmma_gemm_f32<0><<<grid, 256, 0, stream>>>(
        xc, W_xproj, proj, nullptr, NTOK, NPROJ, D_INNER, D_INNER, NPROJ,
        NPROJ);
  }
  // K4: delta = softplus(proj[:, :64] @ W_dt + b_dt)  M=2048 N=2048 K=64
  {
    dim3 grid(D_INNER / GEMM_TN, NTOK / GEMM_TM);
    wmma_gemm_f32<1><<<grid, 256, 0, stream>>>(
        proj, W_dt, delta, b_dt, NTOK, D_INNER, DT_RANK, NPROJ, D_INNER,
        D_INNER);
  }
  // K5: scan -> y (includes * silu(z))
  {
    int nthreads = BATCH * D_INNER;  // 4096
    ssm_scan_kernel<<<nthreads / 256, 256, 0, stream>>>(proj, delta, xc, xz,
                                                        A_log, D_param, y);
  }
  // K6: out = y @ W_out      M=2048 N=1024 K=2048
  {
    dim3 grid(D_MODEL / GEMM_TN, NTOK / GEMM_TM);
    wmma_gemm_f32<0><<<grid, 256, 0, stream>>>(
        y, W_out, out, nullptr, NTOK, D_MODEL, D_INNER, D_INNER, D_MODEL,
        D_MODEL);
  }
}